// GCNEncoder_82566451298969
// MI455X (gfx1250) — compile-verified
//
#include <hip/hip_runtime.h>

typedef __attribute__((ext_vector_type(2))) float v2f;
typedef __attribute__((ext_vector_type(8))) float v8f;

#define D   128   // feature dim (in = hid = out = 128)
#define MT  128   // rows per block (8 waves x 16 rows)

// ---------------------------------------------------------------------------
// degree count: one thread per edge, atomic +1 on dst
// ---------------------------------------------------------------------------
__global__ void k_deg(const long long* __restrict__ ei, long long E,
                      float* __restrict__ deg) {
    long long i = (long long)blockIdx.x * blockDim.x + threadIdx.x;
    if (i < E) {
        int d = (int)ei[E + i];
        atomicAdd(&deg[d], 1.0f);
    }
}

// dinv = rsqrt(deg + 1)   (self-loop contributes +1, so deg+1 >= 1 always)
__global__ void k_dinv(float* __restrict__ deg, int n) {
    int i = blockIdx.x * blockDim.x + threadIdx.x;
    if (i < n) deg[i] = rsqrtf(deg[i] + 1.0f);
}

// ---------------------------------------------------------------------------
// WMMA GEMM: H = (RELU? max(X,0) : X) @ W    [nrows x 128] @ [128 x 128]
// Also writes ACC = H * dinv[row]^2 + bias[col]  (self-loop msg + bias fused)
//
// Per block: 256 threads = 8 waves; wave w covers rows row0..row0+15.
// K accumulated via 32 chained V_WMMA_F32_16X16X4_F32.
// W staged in LDS pre-shuffled into B-fragment order:
//   slot[((kstep*8+nt)*32+lane)*2+j] = W[kstep*4 + 2*(lane/16) + j][nt*16 + lane%16]
// so the inner-loop B load is a lane-contiguous 8B LDS read (conflict-free).
//
// Out-of-range A rows are CLAMPED (not masked): a garbage A row only affects
// the same out-of-range D row, which the epilogue never stores. This keeps
// the inner loop free of EXEC manipulation (WMMA needs EXEC all-ones anyway).
// ---------------------------------------------------------------------------
template <bool RELU>
__global__ __launch_bounds__(256)
void k_gemm(const float* __restrict__ X, const float* __restrict__ W,
            const float* __restrict__ bias, const float* __restrict__ dinv,
            float* __restrict__ H, float* __restrict__ ACC, int nrows) {
    __shared__ float Wf[32 * 8 * 32 * 2];   // 64 KB

    const int t = threadIdx.x;

    // stage + shuffle W (coalesced global reads, one-time scattered LDS writes)
    #pragma unroll
    for (int i = 0; i < 64; ++i) {
        int idx   = i * 256 + t;          // 0..16383
        int k     = idx >> 7;
        int n     = idx & 127;
        int kstep = k >> 2, kr = k & 3;
        int half  = kr >> 1, j = kr & 1;
        int nt    = n >> 4,  m = n & 15;
        int lane  = half * 16 + m;
        Wf[(((kstep << 3) + nt) << 6) + (lane << 1) + j] = W[idx];
    }
    __syncthreads();

    const int wave = t >> 5;
    const int lane = t & 31;
    const int half = lane >> 4;           // 0: lanes 0-15, 1: lanes 16-31
    const int mrow = lane & 15;
    const int row0 = blockIdx.x * MT + wave * 16;
    int arow = row0 + mrow;
    if (arow >= nrows) arow = nrows - 1;  // clamp: safe addr, result discarded
    const float* xrow = X + (long long)arow * D;

    v8f acc[8];
    #pragma unroll
    for (int nt = 0; nt < 8; ++nt) acc[nt] = {};

    #pragma unroll 4
    for (int ks = 0; ks < 32; ++ks) {
        // A fragment (16x4 fp32): lanes 0-15 -> K = 4ks+0,1 ; lanes 16-31 -> K = 4ks+2,3
        int col = (ks << 2) + (half << 1);
        v2f a = *(const v2f*)(xrow + col);
        if (RELU) {
            a.x = fmaxf(a.x, 0.0f);
            a.y = fmaxf(a.y, 0.0f);
        }
        const float* wl = &Wf[(ks << 9) + (lane << 1)];
        #pragma unroll
        for (int nt = 0; nt < 8; ++nt) {
            v2f b = *(const v2f*)(wl + (nt << 6));
            acc[nt] = __builtin_amdgcn_wmma_f32_16x16x4_f32(
                false, a, false, b, (short)0, acc[nt], false, false);
        }
    }

    // epilogue: C/D layout -> VGPR i holds M = i (lanes 0-15) / M = i+8 (lanes 16-31)
    float srow[8];
    #pragma unroll
    for (int i = 0; i < 8; ++i) {
        int r = row0 + i + half * 8;
        int rc = (r < nrows) ? r : (nrows - 1);   // clamped load; store guarded below
        float dv = dinv[rc];
        srow[i] = dv * dv;                        // self-loop norm = 1/deg
    }
    #pragma unroll
    for (int nt = 0; nt < 8; ++nt) {
        int col  = (nt << 4) + mrow;
        float bc = bias[col];
        #pragma unroll
        for (int i = 0; i < 8; ++i) {
            int r = row0 + i + half * 8;
            if (r < nrows) {
                float v = acc[nt][i];
                H  [(long long)r * D + col] = v;
                ACC[(long long)r * D + col] = v * srow[i] + bc;
            }
        }
    }
}

// ---------------------------------------------------------------------------
// edge scatter: one wave (32 lanes) per edge; each lane owns 4 features.
// acc[dst][:] += h[src][:] * dinv[src]*dinv[dst]
// (h, acc and dinv are all L2-resident: ~154MB working set < 192MB L2)
// ---------------------------------------------------------------------------
__global__ void k_scatter(const float* __restrict__ H,
                          const long long* __restrict__ ei,
                          const float* __restrict__ dinv,
                          float* __restrict__ acc, long long E) {
    long long gid = (long long)blockIdx.x * blockDim.x + threadIdx.x;
    long long e   = gid >> 5;
    if (e >= E) return;
    int lane = (int)(gid & 31);
    int s = (int)ei[e];
    int d = (int)ei[E + e];
    float norm = dinv[s] * dinv[d];
    const float4 v = *(const float4*)&H[(long long)s * D + lane * 4];
    float* p = &acc[(long long)d * D + lane * 4];
    atomicAdd(p + 0, v.x * norm);
    atomicAdd(p + 1, v.y * norm);
    atomicAdd(p + 2, v.z * norm);
    atomicAdd(p + 3, v.w * norm);
}

// ---------------------------------------------------------------------------
extern "C" void kernel_launch(void* const* d_in, const int* in_sizes, int n_in,
                              void* d_out, int out_size, void* d_ws, size_t ws_size,
                              hipStream_t stream) {
    const float*     x  = (const float*)d_in[0];
    const long long* ei = (const long long*)d_in[1];
    const float*     W1 = (const float*)d_in[2];
    const float*     b1 = (const float*)d_in[3];
    const float*     W2 = (const float*)d_in[4];
    const float*     b2 = (const float*)d_in[5];
    float* out = (float*)d_out;

    const int       N = in_sizes[0] / D;                 // 100000
    const long long E = (long long)(in_sizes[1] / 2);    // 1600000

    char* ws = (char*)d_ws;
    float* deg = (float*)ws;                                          // N floats (becomes dinv)
    size_t off = (((size_t)N * sizeof(float)) + 255) & ~(size_t)255;
    float* H   = (float*)(ws + off);                                  // N*D floats
    float* ACC = (float*)(ws + off + (size_t)N * D * sizeof(float));  // N*D floats

    // normalization
    hipMemsetAsync(deg, 0, (size_t)N * sizeof(float), stream);
    k_deg<<<(unsigned)((E + 255) / 256), 256, 0, stream>>>(ei, E, deg);
    k_dinv<<<(N + 255) / 256, 256, 0, stream>>>(deg, N);

    const int mblocks = (N + MT - 1) / MT;
    const long long sthreads = E * 32;
    const unsigned sblocks = (unsigned)((sthreads + 255) / 256);

    // layer 1: h1 = x@W1 ; acc = h1*dinv^2 + b1 ; acc += scatter(h1)
    k_gemm<false><<<mblocks, 256, 0, stream>>>(x, W1, b1, deg, H, ACC, N);
    k_scatter<<<sblocks, 256, 0, stream>>>(H, ei, deg, ACC, E);

    // layer 2: h2 = relu(acc)@W2 ; out = h2*dinv^2 + b2 ; out += scatter(h2)
    k_gemm<true><<<mblocks, 256, 0, stream>>>(ACC, W2, b2, deg, H, out, N);
    k_scatter<<<sblocks, 256, 0, stream>>>(H, ei, deg, out, E);
}